// SimpleGCNLayer_59846074303178
// MI455X (gfx1250) — compile-verified
//
#include <hip/hip_runtime.h>

typedef __attribute__((ext_vector_type(2))) float v2f;
typedef __attribute__((ext_vector_type(8))) float v8f;

#define GCN_EPS 1e-8f

// ---------------------------------------------------------------- zero mask
__global__ void gcn_zero_u32(unsigned* __restrict__ p, long n) {
    long i = (long)blockIdx.x * blockDim.x + threadIdx.x;
    long stride = (long)gridDim.x * blockDim.x;
    for (; i < n; i += stride) p[i] = 0u;
}

// ------------------------------------------- scatter edges + self loops (dedup via bitmask)
__global__ void gcn_scatter(const long long* __restrict__ ei, long E, int N,
                            int wordsPerRow, unsigned* __restrict__ mask) {
    long total = E + N;
    long i = (long)blockIdx.x * blockDim.x + threadIdx.x;
    long stride = (long)gridDim.x * blockDim.x;
    for (; i < total; i += stride) {
        int src, dst;
        if (i < E) {
            src = (int)ei[i];        // edge_index[0, e]
            dst = (int)ei[E + i];    // edge_index[1, e]
        } else {
            src = dst = (int)(i - E);  // self loop
        }
        size_t w = (size_t)src * wordsPerRow + (dst >> 5);
        atomicOr(&mask[w], 1u << (dst & 31));
    }
}

// ------------------------------------------- row degrees -> d^-1/2
__global__ void gcn_degrees(const unsigned* __restrict__ mask, int N,
                            int wordsPerRow, float* __restrict__ dinv) {
    int row = blockIdx.x * blockDim.x + threadIdx.x;
    if (row >= N) return;
    const unsigned* p = mask + (size_t)row * wordsPerRow;
    int deg = 0;
    for (int w = 0; w < wordsPerRow; ++w) deg += __popc(p[w]);
    dinv[row] = rsqrtf((float)deg + GCN_EPS);
}

// ------------------------------------------- y[j,:] = dinv[j] * x[j,:]   (D=128, float4 per thread)
__global__ void gcn_scale(const float* __restrict__ x, const float* __restrict__ dinv,
                          float* __restrict__ y, int N) {
    int t = blockIdx.x * blockDim.x + threadIdx.x;   // N*32 threads
    int row = t >> 5;
    int c4 = t & 31;
    if (row >= N) return;
    float4 v = ((const float4*)x)[(size_t)row * 32 + c4];
    float d = dinv[row];
    v.x *= d; v.y *= d; v.z *= d; v.w *= d;
    ((float4*)y)[(size_t)row * 32 + c4] = v;
}

// ------------------------------------------- sparse aggregate: one wave32 per row
__global__ void gcn_aggregate(const unsigned* __restrict__ mask, const float* __restrict__ y,
                              const float* __restrict__ dinv, float* __restrict__ agg,
                              int N, int wordsPerRow) {
    int lane = threadIdx.x & 31;
    int wave = threadIdx.x >> 5;
    int row = blockIdx.x * (blockDim.x >> 5) + wave;
    if (row >= N) return;
    const unsigned* mrow = mask + (size_t)row * wordsPerRow;
    float4 acc = make_float4(0.f, 0.f, 0.f, 0.f);
    for (int w = 0; w < wordsPerRow; ++w) {
        // address is wave-uniform; force the bit loop into SALU
        unsigned m = (unsigned)__builtin_amdgcn_readfirstlane((int)mrow[w]);
        while (m) {
            int b = __builtin_ctz(m);
            m &= m - 1;
            int j = (w << 5) + b;
            float4 v = ((const float4*)y)[(size_t)j * 32 + lane];
            acc.x += v.x; acc.y += v.y; acc.z += v.z; acc.w += v.w;
        }
    }
    float d = dinv[row];
    float4 o = make_float4(acc.x * d, acc.y * d, acc.z * d, acc.w * d);
    ((float4*)agg)[(size_t)row * 32 + lane] = o;
}

// ------------------------------------------- out = agg @ W^T + b  via V_WMMA_F32_16X16X4_F32
// One wave32 per 16x16 output tile. 8 waves/block cover the full Dout=128 (8 N-tiles).
__global__ void gcn_gemm_wmma(const float* __restrict__ agg, const float* __restrict__ W,
                              const float* __restrict__ bias, float* __restrict__ out) {
    const int D = 128;
    int lane = threadIdx.x & 31;
    int wave = threadIdx.x >> 5;       // 0..7 -> N-tile
    int mbase = blockIdx.x * 16;
    int nbase = wave * 16;
    int r  = lane & 15;                // M (for A) / N (for B)
    int kb = (lane >> 4) << 1;         // K sub-offset: lanes 0-15 -> K 0,1 ; lanes 16-31 -> K 2,3

    const float* arow = agg + (size_t)(mbase + r) * D;   // A[m, k]
    const float* wrow = W   + (size_t)(nbase + r) * D;   // B[k, n] = W[n, k]

    v8f c = {0.f, 0.f, 0.f, 0.f, 0.f, 0.f, 0.f, 0.f};
    #pragma unroll
    for (int k0 = 0; k0 < D; k0 += 4) {
        v2f a = *(const v2f*)(arow + k0 + kb);   // 8B aligned: k0+kb is even
        v2f b = *(const v2f*)(wrow + k0 + kb);
        c = __builtin_amdgcn_wmma_f32_16x16x4_f32(
                /*neg_a=*/false, a, /*neg_b=*/false, b,
                /*c_mod=*/(short)0, c, /*reuse_a=*/false, /*reuse_b=*/false);
    }

    // C/D layout: VGPR r holds M=r (lanes 0-15) / M=r+8 (lanes 16-31); N = lane&15
    int n  = lane & 15;
    int mh = (lane >> 4) * 8;
    float bv = bias[nbase + n];
    #pragma unroll
    for (int i = 0; i < 8; ++i) {
        out[(size_t)(mbase + mh + i) * D + nbase + n] = c[i] + bv;
    }
}

// ---------------------------------------------------------------------------
extern "C" void kernel_launch(void* const* d_in, const int* in_sizes, int n_in,
                              void* d_out, int out_size, void* d_ws, size_t ws_size,
                              hipStream_t stream) {
    const float*     x    = (const float*)d_in[0];       // [N,128] f32
    const long long* ei   = (const long long*)d_in[1];   // [2,E] int64
    const float*     W    = (const float*)d_in[2];       // [128,128] f32
    const float*     bias = (const float*)d_in[3];       // [128] f32
    float*           out  = (float*)d_out;               // [N,128] f32

    const int D = 128;
    int  N = in_sizes[0] / D;          // 8192
    long E = in_sizes[1] / 2;          // 262144
    int  wordsPerRow = N / 32;         // 256

    // workspace layout: [mask 8MB][dinv 32KB][y 4MB][agg 4MB]
    size_t maskWords = (size_t)N * wordsPerRow;
    char*     ws   = (char*)d_ws;
    unsigned* mask = (unsigned*)ws;
    float*    dinv = (float*)(ws + maskWords * sizeof(unsigned));
    float*    y    = dinv + N;
    float*    agg  = y + (size_t)N * D;

    // 1) clear adjacency bitmask
    gcn_zero_u32<<<4096, 256, 0, stream>>>(mask, (long)maskWords);

    // 2) scatter edges + self loops (atomicOr => duplicates collapse)
    long total = E + N;
    int sblocks = (int)((total + 255) / 256);
    if (sblocks > 16384) sblocks = 16384;
    gcn_scatter<<<sblocks, 256, 0, stream>>>(ei, E, N, wordsPerRow, mask);

    // 3) degrees -> d^-1/2
    gcn_degrees<<<(N + 255) / 256, 256, 0, stream>>>(mask, N, wordsPerRow, dinv);

    // 4) y = dinv[j] * x
    gcn_scale<<<(N * 32 + 255) / 256, 256, 0, stream>>>(x, dinv, y, N);

    // 5) sparse aggregate (one wave per row, 8 waves per block)
    gcn_aggregate<<<N / 8, 256, 0, stream>>>(mask, y, dinv, agg, N, wordsPerRow);

    // 6) dense linear layer via f32 WMMA, bias fused
    gcn_gemm_wmma<<<N / 16, 256, 0, stream>>>(agg, W, bias, out);
}